// ROIEmbedding1D_23235773071814
// MI455X (gfx1250) — compile-verified
//
#include <hip/hip_runtime.h>

// Multi-scale adaptive max pool (1D SPP), x:[2048,32768] f32 -> out:[2048,63].
// Pure bandwidth-bound streaming reduction (256 MiB read, ~11 us floor @23.3TB/s).
// CDNA5 path: async global->LDS copies (ASYNCcnt) stage each 128KB row in LDS
// (non-temporal hint: 256MiB stream > 192MB L2), then wave32 shuffle reductions.
// No matmul structure -> WMMA not applicable to a max-reduction.

#define ROW_W   32768       // floats per row
#define BLOCK_T 1024        // threads per block (32 wave32s)
#define N_OUT   63          // 1+2+4+8+16+32
#define CPOL_NT 1           // temporal hint TH=NT (bits[2:0]), scope=0

typedef int v4i __attribute__((vector_size(4 * sizeof(int))));
typedef __attribute__((address_space(1))) v4i gv4i;
typedef __attribute__((address_space(3))) v4i lv4i;

__launch_bounds__(BLOCK_T, 2)
__global__ void spp_maxpool_kernel(const float* __restrict__ x,
                                   float* __restrict__ out) {
    __shared__ float buf[ROW_W];      // 128 KB row stage (WGP has 320 KB LDS)
    __shared__ float binmax[32];

    const int row = blockIdx.x;
    const int t   = threadIdx.x;

    // ---- Stage full row into LDS with async copies (16B per lane per round) ----
    const char* grow  = (const char*)(x + (size_t)row * ROW_W);
    char*       lbase = (char*)buf;

#pragma unroll
    for (int k = 0; k < 8; ++k) {
        const int off = (k * BLOCK_T + t) * 16;   // byte offset (coalesced per wave)
#if __has_builtin(__builtin_amdgcn_global_load_async_to_lds_b128)
        __builtin_amdgcn_global_load_async_to_lds_b128(
            (gv4i*)(grow + off),
            (lv4i*)(lbase + off),
            0, CPOL_NT);
#else
        // Fallback: raw CDNA5 mnemonic. LDS addr = low 32 bits of generic ptr.
        unsigned ldsaddr = (unsigned)(size_t)(lbase + off);
        const void* gptr = (const void*)(grow + off);
        asm volatile("global_load_async_to_lds_b128 %0, %1, off th:TH_LOAD_NT"
                     :: "v"(ldsaddr), "v"(gptr) : "memory");
#endif
    }

#if __has_builtin(__builtin_amdgcn_s_wait_asynccnt)
    __builtin_amdgcn_s_wait_asynccnt(0);
#else
    asm volatile("s_wait_asynccnt 0" ::: "memory");
#endif
    __syncthreads();

    // ---- Wave w reduces bin w (1024 floats = 256 float4s) from LDS ----
    const int w = t >> 5;        // bin / wave id: 0..31
    const int l = t & 31;        // lane
    const float4* bufv = (const float4*)buf;

    float m = -__builtin_huge_valf();
#pragma unroll
    for (int k = 0; k < 8; ++k) {
        float4 v = bufv[w * 256 + k * 32 + l];   // ds_load_b128, conflict-free
        m = fmaxf(m, fmaxf(fmaxf(v.x, v.y), fmaxf(v.z, v.w)));
    }
    // wave32 max-reduction
#pragma unroll
    for (int off = 16; off > 0; off >>= 1)
        m = fmaxf(m, __shfl_xor(m, off, 32));
    if (l == 0) binmax[w] = m;
    __syncthreads();

    // ---- Build the 63-output pooling tree from the 32 fine bin maxes ----
    if (t < 32) {
        float* orow = out + (size_t)row * N_OUT;
        // p=32 : out[31..62]
        orow[31 + t] = binmax[t];
        // p=16 : out[15..30]
        if (t < 16)
            orow[15 + t] = fmaxf(binmax[2 * t], binmax[2 * t + 1]);
        // p=8 : out[7..14]
        if (t < 8) {
            float v = binmax[4 * t];
#pragma unroll
            for (int i = 1; i < 4; ++i) v = fmaxf(v, binmax[4 * t + i]);
            orow[7 + t] = v;
        }
        // p=4 : out[3..6]
        if (t < 4) {
            float v = binmax[8 * t];
#pragma unroll
            for (int i = 1; i < 8; ++i) v = fmaxf(v, binmax[8 * t + i]);
            orow[3 + t] = v;
        }
        // p=2 : out[1..2]
        if (t < 2) {
            float v = binmax[16 * t];
#pragma unroll
            for (int i = 1; i < 16; ++i) v = fmaxf(v, binmax[16 * t + i]);
            orow[1 + t] = v;
        }
        // p=1 : out[0]
        if (t == 0) {
            float v = binmax[0];
#pragma unroll
            for (int i = 1; i < 32; ++i) v = fmaxf(v, binmax[i]);
            orow[0] = v;
        }
    }
}

extern "C" void kernel_launch(void* const* d_in, const int* in_sizes, int n_in,
                              void* d_out, int out_size, void* d_ws, size_t ws_size,
                              hipStream_t stream) {
    const float* x   = (const float*)d_in[0];
    float*       out = (float*)d_out;
    const int B = in_sizes[0] / ROW_W;   // 2048 rows, one block per row
    spp_maxpool_kernel<<<B, BLOCK_T, 0, stream>>>(x, out);
}